// GaussianMerge_11897059410348
// MI455X (gfx1250) — compile-verified
//
#include <hip/hip_runtime.h>

#ifndef __has_builtin
#define __has_builtin(x) 0
#endif

#define BLOCK 256
#define EPS 1e-8f

// Native clang vectors (HIP's float4 is a class type -> rejected by
// __builtin_nontemporal_load on the host pass).
typedef float v4f __attribute__((vector_size(16)));
typedef int   v4i __attribute__((vector_size(16)));

// ---------------------------------------------------------------------------
// CDNA5 async global->LDS path (ASYNCcnt). Probe rounds showed the builtins
// are declared with typed pointers:
//   b32 : (addrspace(1) int*,             addrspace(3) int*,             imm, imm)
//   b128: (addrspace(1) int __vector(4)*, addrspace(3) int __vector(4)*, imm, imm)
// ---------------------------------------------------------------------------
#if defined(__gfx1250__) &&                                                   \
    __has_builtin(__builtin_amdgcn_global_load_async_to_lds_b128) &&          \
    __has_builtin(__builtin_amdgcn_global_load_async_to_lds_b32)
#define USE_ASYNC_LDS 1
#else
#define USE_ASYNC_LDS 0
#endif

#if USE_ASYNC_LDS
#define GPTR32(p)  ((__attribute__((address_space(1))) int*)(p))
#define LPTR32(p)  ((__attribute__((address_space(3))) int*)(p))
#define GPTR128(p) ((__attribute__((address_space(1))) v4i*)(p))
#define LPTR128(p) ((__attribute__((address_space(3))) v4i*)(p))
#endif

__device__ __forceinline__ void stage_to_lds(const float* __restrict__ g,
                                             float* s, int total, int tid) {
  const int total4 = total >> 2;  // sections are 16B aligned; n%4==0 in practice
#if USE_ASYNC_LDS
  for (int j = tid; j < total4; j += BLOCK) {
    __builtin_amdgcn_global_load_async_to_lds_b128(
        GPTR128(const_cast<float*>(g) + 4 * j), LPTR128(s + 4 * j), 0, 0);
  }
  for (int j = (total4 << 2) + tid; j < total; j += BLOCK) {
    __builtin_amdgcn_global_load_async_to_lds_b32(
        GPTR32(const_cast<float*>(g) + j), LPTR32(s + j), 0, 0);
  }
#else
  const v4f* g4 = reinterpret_cast<const v4f*>(g);
  v4f* s4 = reinterpret_cast<v4f*>(s);
  for (int j = tid; j < total4; j += BLOCK)
    s4[j] = __builtin_nontemporal_load(g4 + j);
  for (int j = (total4 << 2) + tid; j < total; j += BLOCK)
    s[j] = __builtin_nontemporal_load(g + j);
#endif
}

__device__ __forceinline__ void wait_stage_done() {
#if USE_ASYNC_LDS
#if __has_builtin(__builtin_amdgcn_s_wait_asynccnt)
  __builtin_amdgcn_s_wait_asynccnt(0);
#else
  asm volatile("s_wait_asynccnt 0" ::: "memory");
#endif
#endif
  __syncthreads();  // compiler inserts ds/load waits for the fallback path
}

__global__ __launch_bounds__(BLOCK) void gaussian_merge_kernel(
    const float* __restrict__ loc1, const float* __restrict__ scale1,
    const float* __restrict__ rot1, const float* __restrict__ loc2,
    const float* __restrict__ scale2, const float* __restrict__ rot2,
    float* __restrict__ out_loc,   // [B*3]
    float* __restrict__ out_cov,   // [B*9]
    int B) {
  // 30 floats/element * 256 = 30720 B of LDS; all section sizes are
  // multiples of 4 floats -> every section stays 16B aligned.
  __shared__ __align__(16) float smem[BLOCK * 30];
  float* s_rot1 = smem;                // 9*BLOCK
  float* s_rot2 = smem + 9 * BLOCK;    // 9*BLOCK
  float* s_loc1 = smem + 18 * BLOCK;   // 3*BLOCK
  float* s_loc2 = smem + 21 * BLOCK;   // 3*BLOCK
  float* s_sc1  = smem + 24 * BLOCK;   // 3*BLOCK
  float* s_sc2  = smem + 27 * BLOCK;   // 3*BLOCK

  const int tid = threadIdx.x;
  const int blockStart = blockIdx.x * BLOCK;
  int n = B - blockStart;
  if (n <= 0) return;
  if (n > BLOCK) n = BLOCK;

  // Coalesced AoS -> LDS staging (async DMA on CDNA5 when available).
  stage_to_lds(rot1 + (size_t)blockStart * 9, s_rot1, n * 9, tid);
  stage_to_lds(rot2 + (size_t)blockStart * 9, s_rot2, n * 9, tid);
  stage_to_lds(loc1 + (size_t)blockStart * 3, s_loc1, n * 3, tid);
  stage_to_lds(loc2 + (size_t)blockStart * 3, s_loc2, n * 3, tid);
  stage_to_lds(scale1 + (size_t)blockStart * 3, s_sc1, n * 3, tid);
  stage_to_lds(scale2 + (size_t)blockStart * 3, s_sc2, n * 3, tid);
  wait_stage_done();

  if (tid < n) {
    const int i = blockStart + tid;

    float r1[9], r2[9], sc1[3], sc2[3], l1[3], l2[3];
#pragma unroll
    for (int k = 0; k < 9; ++k) {
      r1[k] = s_rot1[tid * 9 + k];   // stride 9 vs 64 banks -> conflict-free
      r2[k] = s_rot2[tid * 9 + k];
    }
#pragma unroll
    for (int k = 0; k < 3; ++k) {
      l1[k] = s_loc1[tid * 3 + k];
      l2[k] = s_loc2[tid * 3 + k];
      sc1[k] = s_sc1[tid * 3 + k];
      sc2[k] = s_sc2[tid * 3 + k];
    }

    // c = R diag(s) R^T : c[a][b] = sum_j R[a][j]*s[j]*R[b][j]   (symmetric)
    float c1[3][3], c2[3][3];
#pragma unroll
    for (int a = 0; a < 3; ++a) {
#pragma unroll
      for (int b = 0; b <= a; ++b) {
        float acc1 = 0.f, acc2 = 0.f;
#pragma unroll
        for (int j = 0; j < 3; ++j) {
          acc1 = fmaf(r1[a * 3 + j] * sc1[j], r1[b * 3 + j], acc1);
          acc2 = fmaf(r2[a * 3 + j] * sc2[j], r2[b * 3 + j], acc2);
        }
        c1[a][b] = acc1; c1[b][a] = acc1;
        c2[a][b] = acc2; c2[b][a] = acc2;
      }
    }

    // A = c1 + c2 + eps*I (symmetric 3x3)
    const float A00 = c1[0][0] + c2[0][0] + EPS;
    const float A01 = c1[0][1] + c2[0][1];
    const float A02 = c1[0][2] + c2[0][2];
    const float A11 = c1[1][1] + c2[1][1] + EPS;
    const float A12 = c1[1][2] + c2[1][2];
    const float A22 = c1[2][2] + c2[2][2] + EPS;

    // inv(A) via adjugate (symmetric): one reciprocal, no divergence
    float M00 = A11 * A22 - A12 * A12;
    float M01 = A02 * A12 - A01 * A22;
    float M02 = A01 * A12 - A02 * A11;
    float M11 = A00 * A22 - A02 * A02;
    float M12 = A01 * A02 - A00 * A12;
    float M22 = A00 * A11 - A01 * A01;
    const float det = A00 * M00 + A01 * M01 + A02 * M02;
    const float inv = 1.0f / det;
    M00 *= inv; M01 *= inv; M02 *= inv;
    M11 *= inv; M12 *= inv; M22 *= inv;
    const float M[3][3] = {{M00, M01, M02}, {M01, M11, M12}, {M02, M12, M22}};

    // K = c1 * inv(A)
    float K[3][3];
#pragma unroll
    for (int a = 0; a < 3; ++a)
#pragma unroll
      for (int b = 0; b < 3; ++b)
        K[a][b] = fmaf(c1[a][0], M[0][b],
                  fmaf(c1[a][1], M[1][b], c1[a][2] * M[2][b]));

    // loc_new = loc1 + K (loc2 - loc1)
    const float d0 = l2[0] - l1[0], d1 = l2[1] - l1[1], d2 = l2[2] - l1[2];
#pragma unroll
    for (int a = 0; a < 3; ++a) {
      const float v =
          l1[a] + fmaf(K[a][0], d0, fmaf(K[a][1], d1, K[a][2] * d2));
      __builtin_nontemporal_store(v, out_loc + (size_t)i * 3 + a);
    }

    // cov_new = c1 + K * c1
#pragma unroll
    for (int a = 0; a < 3; ++a)
#pragma unroll
      for (int b = 0; b < 3; ++b) {
        const float v = c1[a][b] +
            fmaf(K[a][0], c1[0][b],
            fmaf(K[a][1], c1[1][b], K[a][2] * c1[2][b]));
        __builtin_nontemporal_store(v, out_cov + (size_t)i * 9 + a * 3 + b);
      }
  }
}

extern "C" void kernel_launch(void* const* d_in, const int* in_sizes, int n_in,
                              void* d_out, int out_size, void* d_ws,
                              size_t ws_size, hipStream_t stream) {
  const float* loc1   = (const float*)d_in[0];
  const float* scale1 = (const float*)d_in[1];
  const float* rot1   = (const float*)d_in[2];
  const float* loc2   = (const float*)d_in[3];
  const float* scale2 = (const float*)d_in[4];
  const float* rot2   = (const float*)d_in[5];
  // d_in[6]/d_in[7] (cov1/cov2 placeholders) are recomputed by the reference,
  // so they are never read.

  const int B = in_sizes[0] / 3;
  float* out_loc = (float*)d_out;                  // [B,3] flattened
  float* out_cov = (float*)d_out + (size_t)B * 3;  // [B,3,3] flattened

  const int grid = (B + BLOCK - 1) / BLOCK;
  gaussian_merge_kernel<<<grid, BLOCK, 0, stream>>>(
      loc1, scale1, rot1, loc2, scale2, rot2, out_loc, out_cov, B);
}